// StaticPlusResidualVolumeField_22514218566157
// MI455X (gfx1250) — compile-verified
//
#include <hip/hip_runtime.h>

typedef float v2f __attribute__((ext_vector_type(2)));
typedef float v8f __attribute__((ext_vector_type(8)));

#define BLOCK  128          // 4 waves of 32
#define FPITCH 66           // 64 feature columns + 2 pad floats (bank-conflict-free)

struct Params {
  const float* pts;
  const float* ts;
  const float* statics[3];
  const float* xyts[3];
  const float* xzts[3];
  const float* yzts[3];
  const float* sW[3];
  const float* sb[3];
  const float* rW[3];
  const float* rb[3];
  float* out;
  int n;
};

// Sample 16 channels of grid [16, Dd, H, W] at normalized coords (cx,cy,cz)
// (x->W, y->H, z->Dd), align_corners=True, border clamp. Writes 16 floats to LDS row.
__device__ __forceinline__ void tri_sample16(const float* __restrict__ g,
                                             int W, int H, int Dd,
                                             float cx, float cy, float cz,
                                             float* __restrict__ row, int colBase)
{
  float x = (cx + 1.f) * 0.5f * (float)(W - 1);
  float y = (cy + 1.f) * 0.5f * (float)(H - 1);
  float z = (cz + 1.f) * 0.5f * (float)(Dd - 1);
  float xf = floorf(x), yf = floorf(y), zf = floorf(z);
  float fx = x - xf, fy = y - yf, fz = z - zf;
  int x0 = min(max((int)xf, 0), W - 1);
  int y0 = min(max((int)yf, 0), H - 1);
  int z0 = min(max((int)zf, 0), Dd - 1);
  int x1 = min(x0 + 1, W - 1);
  int y1 = min(y0 + 1, H - 1);
  int z1 = min(z0 + 1, Dd - 1);
  float gx = 1.f - fx, gy = 1.f - fy, gz = 1.f - fz;
  float w000 = gz*gy*gx, w001 = gz*gy*fx, w010 = gz*fy*gx, w011 = gz*fy*fx;
  float w100 = fz*gy*gx, w101 = fz*gy*fx, w110 = fz*fy*gx, w111 = fz*fy*fx;
  int r0 = z0*H, r1 = z1*H;
  int o000 = (r0+y0)*W + x0, o001 = (r0+y0)*W + x1;
  int o010 = (r0+y1)*W + x0, o011 = (r0+y1)*W + x1;
  int o100 = (r1+y0)*W + x0, o101 = (r1+y0)*W + x1;
  int o110 = (r1+y1)*W + x0, o111 = (r1+y1)*W + x1;
  int CH = W*H*Dd;
#pragma unroll
  for (int c = 0; c < 16; ++c) {
    const float* gc = g + c*CH;
    float v = gc[o000]*w000 + gc[o001]*w001 + gc[o010]*w010 + gc[o011]*w011
            + gc[o100]*w100 + gc[o101]*w101 + gc[o110]*w110 + gc[o111]*w111;
    row[colBase + c] = v;
  }
}

__global__ __launch_bounds__(BLOCK)
void svf_fused_kernel(Params prm)
{
  // Per-point 64-wide feature rows (pitch 66 -> ds_load_b64 A-fragments conflict-free)
  __shared__ float s_feat[BLOCK * FPITCH];
  // Combined weight Wc[16 out][64 in], stored as B-matrix pairs:
  // s_B[(p*16 + n)*2 + j] = B[2p+j][n] = Wc[n][2p+j]
  __shared__ float s_B[1024];
  __shared__ float s_bias[16];

  const int tid  = (int)threadIdx.x;
  const int lane = tid & 31;
  const int wave = tid >> 5;
  const int n    = prm.n;
  const int blockStart = (int)blockIdx.x * BLOCK;
  const bool fullBlock = (blockStart + BLOCK) <= n;   // uniform: true for all but last block

  // Gather index: clamp tail lanes (duplicate work, masked at store) so EXEC
  // stays all-ones for the WMMA section (ISA requirement).
  int pcIdx = min(blockStart + tid, n - 1);

  // Point / time normalization (matches reference exactly)
  float px = prm.pts[pcIdx*3 + 0];
  float py = prm.pts[pcIdx*3 + 1];
  float pz = prm.pts[pcIdx*3 + 2];
  float tt = prm.ts[pcIdx];
  const float kB = 1.3f;                    // BOUNDS
  const float invB = 2.f / (-kB - kB);
  float xn = (px - kB)*invB - 1.f;
  float yn = (py - kB)*invB - 1.f;
  float zn = (pz - kB)*invB - 1.f;
  float tn = fminf(fmaxf(tt, 0.f), 1.f)*2.f - 1.f;
  xn = fminf(fmaxf(xn, -1.f), 1.f);
  yn = fminf(fmaxf(yn, -1.f), 1.f);
  zn = fminf(fmaxf(zn, -1.f), 1.f);

  const int SRES[3] = {64, 128, 128};   // static grid D=H=W
  const int RRES[3] = {32, 64, 64};     // residual spatial res (rx=ry=rz)
  const int TRES[3] = {12, 25, 25};     // residual time res

  for (int lvl = 0; lvl < 3; ++lvl) {
    // ---- stage fused weight [16 x 64] into LDS in B-fragment pair layout ----
    const float* sW = prm.sW[lvl];
    const float* rW = prm.rW[lvl];
    for (int i = tid; i < 1024; i += BLOCK) {
      int j  = i & 1;          // element within pair
      int nn = (i >> 1) & 15;  // output channel (N)
      int pp = i >> 5;         // K-pair index
      int k  = 2*pp + j;       // K in 0..63
      s_B[i] = (k < 16) ? sW[nn*16 + k] : rW[nn*48 + (k - 16)];
    }
    if (tid < 16) s_bias[tid] = prm.sb[lvl][tid] + prm.rb[lvl][tid];

    // ---- gather: one point per lane, 64 features -> LDS row ----
    float* row = &s_feat[tid * FPITCH];
    int S = SRES[lvl], R = RRES[lvl], T = TRES[lvl];
    tri_sample16(prm.statics[lvl], S, S, S, xn, yn, zn, row, 0);   // static
    tri_sample16(prm.xyts[lvl],    R, R, T, xn, yn, tn, row, 16);  // f1
    tri_sample16(prm.xzts[lvl],    R, R, T, xn, zn, tn, row, 32);  // f2
    tri_sample16(prm.yzts[lvl],    R, R, T, yn, zn, tn, row, 48);  // f3

    __syncthreads();

    // ---- WMMA: [16 pts x 64] @ [64 x 16] via 16 chained F32 16x16x4 ----
    int m     = lane & 15;   // A row (M) / B col (N)
    int khalf = lane >> 4;   // 0: K pair {4s,4s+1}, 1: {4s+2,4s+3}
    float bias = s_bias[m];
#pragma unroll
    for (int tile = 0; tile < 2; ++tile) {
      int rowBase = wave*32 + tile*16;
      v8f acc;
#pragma unroll
      for (int r = 0; r < 8; ++r) acc[r] = bias;   // C init = fused bias (per N)

      const float* arow  = &s_feat[(rowBase + m)*FPITCH + 2*khalf];
      const float* bbase = &s_B[khalf*32 + m*2];
#pragma unroll
      for (int s = 0; s < 16; ++s) {
        v2f a = *(const v2f*)(arow  + 4*s);    // A[M=m][K=4s+2*khalf .. +1]
        v2f b = *(const v2f*)(bbase + 64*s);   // B[K=4s+2*khalf..][N=m]
        acc = __builtin_amdgcn_wmma_f32_16x16x4_f32(
            /*neg_a=*/false, a, /*neg_b=*/false, b,
            /*c_mod=*/(short)0, acc, /*reuse_a=*/false, /*reuse_b=*/false);
      }

      // D layout: VGPR r holds M=r (lanes 0-15) / M=r+8 (lanes 16-31), N=lane&15.
      // Lanes 0-15 / 16-31 each write a contiguous 64B segment per r (coalesced).
      int row0 = blockStart + rowBase + khalf*8;
      float* outp = prm.out + (size_t)row0*48 + lvl*16 + m;
      if (fullBlock) {
#pragma unroll
        for (int r = 0; r < 8; ++r) outp[r*48] = acc[r];   // unconditional fast path
      } else {
#pragma unroll
        for (int r = 0; r < 8; ++r) {
          if (row0 + r < n) outp[r*48] = acc[r];           // tail block only
        }
      }
    }
    __syncthreads();   // protect s_feat / s_B before next level restages
  }
}

extern "C" void kernel_launch(void* const* d_in, const int* in_sizes, int n_in,
                              void* d_out, int out_size, void* d_ws, size_t ws_size,
                              hipStream_t stream)
{
  Params prm;
  prm.pts = (const float*)d_in[0];
  prm.ts  = (const float*)d_in[1];
  for (int l = 0; l < 3; ++l) {
    prm.statics[l] = (const float*)d_in[2  + l];
    prm.xyts[l]    = (const float*)d_in[5  + l];
    prm.xzts[l]    = (const float*)d_in[8  + l];
    prm.yzts[l]    = (const float*)d_in[11 + l];
    prm.sW[l]      = (const float*)d_in[14 + l];
    prm.sb[l]      = (const float*)d_in[17 + l];
    prm.rW[l]      = (const float*)d_in[20 + l];
    prm.rb[l]      = (const float*)d_in[23 + l];
  }
  prm.out = (float*)d_out;
  prm.n   = in_sizes[1];   // timestamps count == N

  int nb = (prm.n + BLOCK - 1) / BLOCK;
  svf_fused_kernel<<<nb, BLOCK, 0, stream>>>(prm);
}